// GFlowNetActor_40312563040654
// MI455X (gfx1250) — compile-verified
//
#include <hip/hip_runtime.h>
#include <hip/hip_bf16.h>
#include <float.h>
#include <math.h>
#include <stdint.h>

// ---------------------------------------------------------------------------
// GFlowNet actor log-prob kernel for MI455X (gfx1250).
// Memory-bound segmented top-K + logsumexp; no matmul structure -> WMMA N/A.
// CDNA5 features: wave32 sizing, GLOBAL_LOAD_ASYNC_TO_LDS_B32/B128 staging,
// s_wait_asynccnt, LDS atomics, large-LDS WGP (55 KB static/block).
// ---------------------------------------------------------------------------

#define THREADS 256
#define TOPK    32
#define TILE    4096          // floats staged per async tile (16 KB)
#define LEN_MAX 32768         // segment-length cap (mean 16384, ~128 sigma)
#define TIE_CAP 128

static constexpr float kInvTemp = 1.0f;      // TEMP = 1.0
static constexpr float kNegInf  = -FLT_MAX;  // jnp.finfo(f32).min

// ---- CDNA5 async-load-to-LDS (guarded; sync fallback keeps compile clean) --
#if defined(__has_builtin)
# if __has_builtin(__builtin_amdgcn_global_load_async_to_lds_b32)
#  define ATHENA_ASYNC_LDS 1
# endif
# if __has_builtin(__builtin_amdgcn_global_load_async_to_lds_b128)
#  define ATHENA_ASYNC_B128 1
# endif
# if __has_builtin(__builtin_amdgcn_s_wait_asynccnt)
#  define ATHENA_WAIT_ASYNC_BUILTIN 1
# endif
#endif

typedef __attribute__((address_space(1))) int athena_gint;  // "__device__ int"
typedef __attribute__((address_space(3))) int athena_lint;  // "__shared__ int"
typedef int athena_v4i __attribute__((vector_size(4 * sizeof(int))));
typedef __attribute__((address_space(1))) athena_v4i athena_gv4i;
typedef __attribute__((address_space(3))) athena_v4i athena_lv4i;

__device__ __forceinline__ void async_copy_f32(float* lds_dst, const float* g_src) {
#ifdef ATHENA_ASYNC_LDS
  __builtin_amdgcn_global_load_async_to_lds_b32(
      (athena_gint*)(void*)g_src,
      (athena_lint*)(void*)lds_dst,
      /*imm offset*/0, /*cpol*/0);
#else
  *lds_dst = *g_src;
#endif
}

__device__ __forceinline__ void async_copy_b128(const void* g_src, void* lds_dst) {
#if defined(ATHENA_ASYNC_LDS) && defined(ATHENA_ASYNC_B128)
  __builtin_amdgcn_global_load_async_to_lds_b128(
      (athena_gv4i*)(void*)g_src,
      (athena_lv4i*)lds_dst,
      /*imm offset*/0, /*cpol*/0);
#else
  *(athena_v4i*)lds_dst = *(const athena_v4i*)g_src;
#endif
}

__device__ __forceinline__ void wait_async_copies() {
#ifdef ATHENA_ASYNC_LDS
# ifdef ATHENA_WAIT_ASYNC_BUILTIN
  __builtin_amdgcn_s_wait_asynccnt(0);
# else
  asm volatile("s_wait_asynccnt 0" ::: "memory");
# endif
#endif
}

// Stage tc floats from gsrc into tile[pad .. pad+tc); returns the phase pad
// (0..3 floats) chosen so body copies are 16B-aligned on BOTH sides -> B128.
__device__ __forceinline__ int stage_tile_f32(float* tile, const float* gsrc,
                                              int tc, int tid) {
  const int pad = (int)((((uintptr_t)gsrc) >> 2) & 3u);
#if defined(ATHENA_ASYNC_LDS) && defined(ATHENA_ASYNC_B128)
  int body_start = (4 - pad) & 3;
  if (body_start > tc) body_start = tc;
  const int body_end = body_start + ((tc - body_start) & ~3);
  if (tid < body_start)                      // head: <=3 elements
    async_copy_f32(&tile[pad + tid], gsrc + tid);
  for (int j = body_start + 4 * tid; j < body_end; j += 4 * THREADS)
    async_copy_b128(gsrc + j, &tile[pad + j]);
  { int j = body_end + tid; if (j < tc) async_copy_f32(&tile[pad + j], gsrc + j); }
#else
  for (int j = tid; j < tc; j += THREADS)
    async_copy_f32(&tile[pad + j], gsrc + j);
#endif
  return pad;
}

// Stage n mask bytes from gsrc into ldsb[pad .. pad+n); 16B-aligned B128 body.
__device__ __forceinline__ int stage_bytes(unsigned char* ldsb,
                                           const unsigned char* gsrc,
                                           int n, int tid) {
  const int pad = (int)(((uintptr_t)gsrc) & 15u);
#if defined(ATHENA_ASYNC_LDS) && defined(ATHENA_ASYNC_B128)
  int body_start = (16 - pad) & 15;
  if (body_start > n) body_start = n;
  const int body_end = body_start + ((n - body_start) & ~15);
  if (tid < body_start) ldsb[pad + tid] = gsrc[tid];          // head: <=15 bytes
  for (int j = body_start + 16 * tid; j < body_end; j += 16 * THREADS)
    async_copy_b128(gsrc + j, &ldsb[pad + j]);
  for (int j = body_end + tid; j < n; j += THREADS) ldsb[pad + j] = gsrc[j];
#else
  for (int j = tid; j < n; j += THREADS) ldsb[pad + j] = gsrc[j];
#endif
  return pad;
}

// Monotonic float -> uint key; 0 reserved for "not a candidate".
__device__ __forceinline__ unsigned int make_key(float sc, unsigned char m) {
  if (!m) return 0u;
  unsigned int u = __float_as_uint(sc);
  return (u & 0x80000000u) ? ~u : (u | 0x80000000u);
}

// ---------------------------------------------------------------------------
// Kernel A: segment boundaries. seg_start[g] = lower_bound(edge_batch, g).
// ---------------------------------------------------------------------------
__global__ void seg_bounds_kernel(const long long* __restrict__ eb,
                                  int* __restrict__ seg_start,
                                  int n_edges, int ng) {
  int g = blockIdx.x * blockDim.x + threadIdx.x;
  if (g > ng) return;
  int lo = 0, hi = n_edges;
  while (lo < hi) {
    int mid = (lo + hi) >> 1;
    if (eb[mid] < (long long)g) lo = mid + 1; else hi = mid;
  }
  seg_start[g] = lo;
}

// ---------------------------------------------------------------------------
// Kernel B: one block per graph. Radix-select threshold, exact tie handling,
// logsumexp with stop action, final write.
// ---------------------------------------------------------------------------
__global__ __launch_bounds__(THREADS)
void actor_kernel(const float* __restrict__ edge_logits,
                  const float* __restrict__ edge_scores,
                  const float* __restrict__ stop_logits,
                  const unsigned char* __restrict__ cand_mask,
                  const int* __restrict__ seg_start,
                  float* __restrict__ out) {
  __shared__ unsigned int hist[256];
  __shared__ unsigned int keepbits[LEN_MAX / 32];
  __shared__ __align__(16) float         tile_scores[TILE + 4];
  __shared__ __align__(16) unsigned char lds_mask[LEN_MAX + 16];
  __shared__ float        kept_logits[TOPK * 2];
  __shared__ unsigned int tie_pos[TIE_CAP];
  __shared__ unsigned int sh_ncand, sh_nkept, sh_ntie, sh_want, sh_thr;
  __shared__ int          sh_keepall;
  __shared__ float        sh_denom;

  const int g   = blockIdx.x;
  const int tid = threadIdx.x;
  const int s   = seg_start[g];
  const int e   = seg_start[g + 1];
  const int len = e - s;
  const int lenc = (len > LEN_MAX) ? LEN_MAX : len;

  if (tid == 0) { sh_ncand = 0; sh_nkept = 0; sh_ntie = 0; }
  for (int w = tid; w < LEN_MAX / 32; w += THREADS) keepbits[w] = 0;
  hist[tid] = 0;

  // Stage the whole segment's candidate mask once (reused by all 5 scans).
  const int padm = stage_bytes(lds_mask, cand_mask + s, lenc, tid);
  __syncthreads();

  // ---- Pass 1: histogram of top 8-bit digit + candidate count -------------
  {
    unsigned int myc = 0;
    for (int tb = 0; tb < lenc; tb += TILE) {
      int tc = min(TILE, lenc - tb);
      int pad = stage_tile_f32(tile_scores, edge_scores + s + tb, tc, tid);
      wait_async_copies();
      __syncthreads();
      for (int j = tid; j < tc; j += THREADS) {
        unsigned int k = make_key(tile_scores[pad + j], lds_mask[padm + tb + j]);
        if (k) { atomicAdd(&hist[k >> 24], 1u); ++myc; }
      }
      __syncthreads();
    }
    if (myc) atomicAdd(&sh_ncand, myc);
  }
  __syncthreads();

  if (tid == 0) {
    if (sh_ncand <= TOPK) {           // keep every candidate
      sh_keepall = 1; sh_thr = 0u; sh_want = 0u;
    } else {
      sh_keepall = 0;
      unsigned int want = TOPK, cum = 0; int b;
      for (b = 255; b >= 0; --b) {
        unsigned int c = hist[b];
        if (cum + c >= want) break;
        cum += c;
      }
      if (b < 0) b = 0;
      sh_thr  = ((unsigned int)b) << 24;
      sh_want = want - cum;
    }
  }
  __syncthreads();

  // ---- Passes 2..4: refine remaining digits -------------------------------
  if (!sh_keepall) {
    const int          shifts[3]  = {16, 8, 0};
    const unsigned int himask[3]  = {0xFF000000u, 0xFFFF0000u, 0xFFFFFF00u};
    for (int p = 0; p < 3; ++p) {
      const int shift = shifts[p];
      const unsigned int hm = himask[p];
      const unsigned int prefix = sh_thr;
      hist[tid] = 0;
      __syncthreads();
      for (int tb = 0; tb < lenc; tb += TILE) {
        int tc = min(TILE, lenc - tb);
        int pad = stage_tile_f32(tile_scores, edge_scores + s + tb, tc, tid);
        wait_async_copies();
        __syncthreads();
        for (int j = tid; j < tc; j += THREADS) {
          unsigned int k = make_key(tile_scores[pad + j], lds_mask[padm + tb + j]);
          if (k && ((k ^ prefix) & hm) == 0u)
            atomicAdd(&hist[(k >> shift) & 0xFFu], 1u);
        }
        __syncthreads();
      }
      if (tid == 0) {
        unsigned int want = sh_want, cum = 0; int b;
        for (b = 255; b >= 0; --b) {
          unsigned int c = hist[b];
          if (cum + c >= want) break;
          cum += c;
        }
        if (b < 0) b = 0;
        sh_thr  = sh_thr | (((unsigned int)b) << shift);
        sh_want = want - cum;
      }
      __syncthreads();
    }
  }

  const unsigned int thr        = sh_thr;
  const unsigned int tiesToKeep = sh_want;
  const int          keepall    = sh_keepall;

  // ---- Pass D1: mark definite keeps, collect ties -------------------------
  for (int tb = 0; tb < lenc; tb += TILE) {
    int tc = min(TILE, lenc - tb);
    int pad = stage_tile_f32(tile_scores, edge_scores + s + tb, tc, tid);
    wait_async_copies();
    __syncthreads();
    for (int j = tid; j < tc; j += THREADS) {
      int pos = tb + j;
      unsigned int k = make_key(tile_scores[pad + j], lds_mask[padm + pos]);
      bool def = keepall ? (k != 0u) : (k > thr);
      if (def) {
        atomicOr(&keepbits[pos >> 5], 1u << (pos & 31));
      } else if (!keepall && k == thr && k != 0u) {
        unsigned int slot = atomicAdd(&sh_ntie, 1u);
        if (slot < TIE_CAP) tie_pos[slot] = (unsigned int)pos;
      }
    }
    __syncthreads();
  }

  // ---- Ties: stable order = ascending index (matches jax stable argsort) --
  if (tid == 0 && !keepall && tiesToKeep > 0) {
    unsigned int nt = sh_ntie; if (nt > TIE_CAP) nt = TIE_CAP;
    for (unsigned int i = 1; i < nt; ++i) {            // insertion sort (tiny)
      unsigned int v = tie_pos[i]; int j = (int)i - 1;
      while (j >= 0 && tie_pos[j] > v) { tie_pos[j + 1] = tie_pos[j]; --j; }
      tie_pos[j + 1] = v;
    }
    unsigned int take = (tiesToKeep < nt) ? tiesToKeep : nt;
    for (unsigned int i = 0; i < take; ++i) {
      unsigned int pos = tie_pos[i];
      keepbits[pos >> 5] |= 1u << (pos & 31);
    }
  }
  __syncthreads();

  // ---- Pass D2: gather kept logits (<= 32 scattered reads) ----------------
  {
    const int nwords = (lenc + 31) >> 5;
    for (int w = tid; w < nwords; w += THREADS) {
      unsigned int bits = keepbits[w];
      while (bits) {
        int b = __ffs(bits) - 1;
        bits &= bits - 1u;
        int pos = (w << 5) + b;
        unsigned int slot = atomicAdd(&sh_nkept, 1u);
        if (slot < TOPK * 2) kept_logits[slot] = edge_logits[s + pos];
      }
    }
  }
  __syncthreads();

  // ---- Logsumexp over kept + stop action ----------------------------------
  if (tid == 0) {
    unsigned int n = sh_nkept; if (n > TOPK * 2) n = TOPK * 2;
    float lse;
    if (n == 0u) {
      lse = kNegInf;                      // matches -FLT_MAX + log(cnt) == -FLT_MAX
    } else {
      float mx = kNegInf;
      for (unsigned int i = 0; i < n; ++i) mx = fmaxf(mx, kept_logits[i] * kInvTemp);
      float sum = 0.0f;
      for (unsigned int i = 0; i < n; ++i) sum += expf(kept_logits[i] * kInvTemp - mx);
      lse = logf(fmaxf(sum, FLT_EPSILON)) + mx;
    }
    float st = stop_logits[g] * kInvTemp;
    float m2 = fmaxf(lse, st);
    float d  = fabsf(lse - st);
    sh_denom = m2 + log1pf(expf(-d));     // logaddexp; exp(-huge)=0 handles -inf side
  }
  __syncthreads();

  // ---- Final coalesced write ----------------------------------------------
  const float denom = sh_denom;
  for (int pos = tid; pos < lenc; pos += THREADS) {
    unsigned int bit = (keepbits[pos >> 5] >> (pos & 31)) & 1u;
    float lt = edge_logits[s + pos] * kInvTemp;
    out[s + pos] = bit ? (lt - denom) : kNegInf;
  }
  for (int pos = lenc + tid; pos < len; pos += THREADS)   // (unreachable in practice)
    out[s + pos] = kNegInf;
}

// ---------------------------------------------------------------------------
extern "C" void kernel_launch(void* const* d_in, const int* in_sizes, int n_in,
                              void* d_out, int out_size, void* d_ws, size_t ws_size,
                              hipStream_t stream) {
  (void)n_in; (void)out_size; (void)ws_size;
  const float*         edge_logits = (const float*)d_in[0];
  const float*         edge_scores = (const float*)d_in[1];
  const float*         stop_logits = (const float*)d_in[2];
  const long long*     edge_batch  = (const long long*)d_in[3];     // int64
  const unsigned char* cand_mask   = (const unsigned char*)d_in[4]; // bool bytes
  float*               out         = (float*)d_out;

  const int n_edges = in_sizes[0];
  const int ng      = in_sizes[2];

  int* seg_start = (int*)d_ws;   // (ng + 1) ints

  seg_bounds_kernel<<<(ng + 1 + 255) / 256, 256, 0, stream>>>(
      edge_batch, seg_start, n_edges, ng);
  actor_kernel<<<ng, THREADS, 0, stream>>>(
      edge_logits, edge_scores, stop_logits, cand_mask, seg_start, out);
}